// KPConvFPN_56831007261091
// MI455X (gfx1250) — compile-verified
//
#include <hip/hip_runtime.h>

typedef __attribute__((ext_vector_type(2))) float v2f;
typedef __attribute__((ext_vector_type(4))) float v4f;
typedef __attribute__((ext_vector_type(8))) float v8f;

namespace {
constexpr int   BATCH = 4;
constexpr int   NQ    = 16384;   // query points per batch
constexpr int   MS    = 16384;   // support points per batch
constexpr int   KNB   = 16;      // neighbors
constexpr int   P     = 15;      // kernel points
constexpr int   CIN   = 64;
constexpr int   COUT  = 128;
constexpr float INV_SIGMA = 1.0f / 0.03f;
constexpr int   TQ    = 16;      // queries per block (one WMMA M-tile)
constexpr int   WFS   = 964;     // padded A-row stride (floats): 4q-bank pattern -> conflict-free b64
constexpr int   KKTOT = P * CIN; // 960 = GEMM K
}

__global__ __launch_bounds__(256)
void kpconv_fused(const float* __restrict__ query,
                  const float* __restrict__ support,
                  const float* __restrict__ feats,
                  const int*   __restrict__ nbridx,
                  const float* __restrict__ weights,
                  const float* __restrict__ bias,
                  const float* __restrict__ kpts,
                  float* __restrict__ out)
{
  __shared__ __align__(16) float arena[TQ * WFS];   // phase1: kw+relp ; phase2: 16x960 A-tile
  __shared__ float qc[TQ * 3];
  __shared__ float kp_s[P * 3];
  __shared__ int   idx_s[TQ * KNB];
  __shared__ int   nzmask[TQ];
  __shared__ float invcnt[TQ];

  float* kw   = arena;            // TQ*KNB*16 = 4096 floats (p-stride padded to 16)
  float* relp = arena + 4096;     // TQ*KNB*3  = 768 floats

  const int tid = threadIdx.x;
  const int blk = blockIdx.x;
  const int b   = blk / (NQ / TQ);
  const int n0  = (blk % (NQ / TQ)) * TQ;

  // ---- phase 0: stage geometry / indices ----
  if (tid < TQ * 3)  qc[tid]   = query[(size_t)(b * NQ + n0) * 3 + tid];
  if (tid < P * 3)   kp_s[tid] = kpts[tid];
  if (tid < TQ * KNB) {
    int id = nbridx[(size_t)(b * NQ + n0) * KNB + tid];
    id = id < 0 ? 0 : (id > MS - 1 ? MS - 1 : id);
    idx_s[tid] = id;
  }
  if (tid < TQ) nzmask[tid] = 0;
  __syncthreads();

  for (int j = tid; j < TQ * KNB * 3; j += 256) {
    int qk = j / 3, d = j % 3;
    relp[j] = support[(size_t)(b * MS + idx_s[qk]) * 3 + d] - qc[(qk / KNB) * 3 + d];
  }
  __syncthreads();

  // kernel correlation weights: kw[q][k][p] = max(1 - |rel - kp|/sigma, 0)
  for (int j = tid; j < TQ * KNB * P; j += 256) {
    int qk = j / P, p = j % P;
    float dx = relp[qk * 3 + 0] - kp_s[p * 3 + 0];
    float dy = relp[qk * 3 + 1] - kp_s[p * 3 + 1];
    float dz = relp[qk * 3 + 2] - kp_s[p * 3 + 2];
    float d2 = fmaf(dx, dx, fmaf(dy, dy, dz * dz));
    float dist = sqrtf(fmaxf(d2, 1e-10f));
    kw[qk * 16 + p] = fmaxf(1.0f - dist * INV_SIGMA, 0.0f);
  }
  __syncthreads();

  // ---- phase 1: weighted features; one query x 4 contiguous channels per thread ----
  // b128 gathers: half-wave reads one 256B feature row; v_pk_fma accumulation.
  const int q  = tid >> 4;          // 0..15
  const int c4 = (tid & 15) * 4;    // channel group
  v4f acc4[P];
  #pragma unroll
  for (int p = 0; p < P; ++p) acc4[p] = (v4f){0.f, 0.f, 0.f, 0.f};
  int mymask = 0;

  for (int k = 0; k < KNB; ++k) {
    const v4f f = *(const v4f*)(feats + (size_t)(b * MS + idx_s[q * KNB + k]) * CIN + c4);
    if (f.x != 0.0f || f.y != 0.0f || f.z != 0.0f || f.w != 0.0f) mymask |= (1 << k);
    const float* kwr = kw + (q * KNB + k) * 16;
    #pragma unroll
    for (int p = 0; p < P; ++p) {
      const float w = kwr[p];
      acc4[p].x = fmaf(w, f.x, acc4[p].x);
      acc4[p].y = fmaf(w, f.y, acc4[p].y);
      acc4[p].z = fmaf(w, f.z, acc4[p].z);
      acc4[p].w = fmaf(w, f.w, acc4[p].w);
    }
  }
  __syncthreads();   // all kw reads complete -> arena reusable as A-tile

  atomicOr(&nzmask[q], mymask);
  #pragma unroll
  for (int p = 0; p < P; ++p)
    *(v4f*)(arena + q * WFS + p * CIN + c4) = acc4[p];
  __syncthreads();
  if (tid < TQ) {
    int cnt = __popc(nzmask[tid]);
    invcnt[tid] = 1.0f / (float)(cnt < 1 ? 1 : cnt);
  }
  __syncthreads();

  // ---- phase 2: (16 x 960) @ (960 x 16) per wave via v_wmma_f32_16x16x4_f32 ----
  const int wv   = tid >> 5;       // wave 0..7 -> channel tile
  const int lane = tid & 31;
  const int lm   = lane & 15;
  const int hi   = lane >> 4;      // 0: K={0,1}, 1: K={2,3} (ISA 16x4 A / mirrored B layout)
  const int co   = wv * 16 + lm;

  v8f acc8 = {0.f, 0.f, 0.f, 0.f, 0.f, 0.f, 0.f, 0.f};
  const float* arow = arena + lm * WFS + hi * 2;                // A: row M=lm
  const float* wcol = weights + (size_t)(hi * 2) * COUT + co;   // B: col N=co

  #pragma unroll 4
  for (int kk = 0; kk < KKTOT; kk += 4) {
    v2f a;
    a.x = arow[kk];
    a.y = arow[kk + 1];
    v2f bb;
    bb.x = wcol[(size_t)kk * COUT];
    bb.y = wcol[(size_t)(kk + 1) * COUT];
    acc8 = __builtin_amdgcn_wmma_f32_16x16x4_f32(false, a, false, bb,
                                                 (short)0, acc8, false, false);
  }

  // ---- epilogue: D tile -> out, normalized + bias ----
  const float bco = bias[co];
  #pragma unroll
  for (int r = 0; r < 8; ++r) {
    int qq = r + hi * 8;           // D layout: VGPR r holds M=r (lanes 0-15) / r+8 (lanes 16-31)
    out[(size_t)(b * NQ + n0 + qq) * COUT + co] = acc8[r] * invcnt[qq] + bco;
  }
}

extern "C" void kernel_launch(void* const* d_in, const int* in_sizes, int n_in,
                              void* d_out, int out_size, void* d_ws, size_t ws_size,
                              hipStream_t stream) {
  const float* query   = (const float*)d_in[0];
  const float* support = (const float*)d_in[1];
  const float* feats   = (const float*)d_in[2];
  const int*   nbr     = (const int*)d_in[3];
  const float* weights = (const float*)d_in[4];
  const float* bias    = (const float*)d_in[5];
  const float* kpts    = (const float*)d_in[6];
  float* outp = (float*)d_out;

  dim3 grid(BATCH * NQ / TQ);   // 4096 blocks, one 16-query M-tile each
  kpconv_fused<<<grid, 256, 0, stream>>>(query, support, feats, nbr,
                                         weights, bias, kpts, outp);
}